// Correlation_87737591922768
// MI455X (gfx1250) — compile-verified
//
#include <hip/hip_runtime.h>

typedef float v2f __attribute__((ext_vector_type(2)));
typedef float v8f __attribute__((ext_vector_type(8)));

#define D_DISP 40
#define NJ     81           // 2*D+1 displacements
#define C_CH   128
#define H_DIM  96
#define W_DIM  320
#define WCHUNK 64           // w columns per block (4 waves x 16)
#define SPAN   144          // WCHUNK + 2*D_DISP : f2p columns a block touches
#define KSTEP  16           // channels staged per iteration

#if __has_builtin(__builtin_amdgcn_global_load_async_to_lds_b32) && \
    __has_builtin(__builtin_amdgcn_s_wait_asynccnt)
#define USE_ASYNC_LDS 1
#else
#define USE_ASYNC_LDS 0
#endif

typedef __attribute__((address_space(1))) int as1_int;
typedef __attribute__((address_space(3))) int as3_int;

// One block = 4 wave32 = 128 threads, handles (b, h, 64-wide w chunk).
// Each wave owns a 16-wide w tile and computes six 16x16 f32 WMMA C-tiles
// (f2p columns w0 .. w0+95), K = 128 channels, double-buffered LDS staging.
__global__ __launch_bounds__(128) void corr_wmma_kernel(
    const float* __restrict__ f1,
    const float* __restrict__ f2,
    float* __restrict__ out)
{
    const int wchunk = blockIdx.x;   // 0..4
    const int h      = blockIdx.y;   // 0..95
    const int b      = blockIdx.z;   // 0..7

    const int tid   = threadIdx.x;
    const int lane  = tid & 31;
    const int wave  = tid >> 5;          // 0..3
    const int mn    = lane & 15;         // M (A/D rows) or N (B/D cols)
    const int lhalf = (lane < 16) ? 0 : 1;

    const int HW = H_DIM * W_DIM;
    const long rowbase = ((long)b * C_CH) * HW + (long)h * W_DIM; // [b, 0, h, 0]

    const int w0  = wchunk * WCHUNK + wave * 16;  // this wave's w-tile start
    const int x0b = wchunk * WCHUNK;              // block f2p span start

    // Double-buffered f2p staging: [buf][sub-chunk of 4 c][x*4 + cc] so a
    // ds b64 read at (x*4 + {0|2}) yields the B-fragment VGPR pair directly.
    __shared__ float sB[2][4][4 * SPAN];
    // output transpose tile: rows j (81) x 64 w, stride 65 avoids bank conflicts
    __shared__ float sOut[NJ * 65];

    // Pre-zero sB: async staging EXEC-masks out-of-range (padding) lanes, and
    // those pad slots are identical every iteration -> they must start at 0.
    for (int e = tid; e < 2 * 4 * 4 * SPAN; e += 128)
        ((float*)sB)[e] = 0.0f;
    __syncthreads();

    // stage channels [k0, k0+16) of the block's f2p span into sB[bufsel];
    // wave 'wave' owns channels k0 + wave*4 + cc  -> sub-chunk index == wave
    auto stage = [&](int bufsel, int k0) {
#pragma unroll
        for (int cc = 0; cc < 4; ++cc) {
            const int c = k0 + wave * 4 + cc;
            const float* __restrict__ src = f2 + rowbase + (long)c * HW;
#pragma unroll
            for (int it = 0; it < 5; ++it) {
                const int xx = lane + it * 32;
                if (xx < SPAN) {
                    const int m = x0b + xx - D_DISP;   // unpadded f2 column
#if USE_ASYNC_LDS
                    if (m >= 0 && m < W_DIM) {
                        __builtin_amdgcn_global_load_async_to_lds_b32(
                            (as1_int*)(src + m),
                            (as3_int*)&sB[bufsel][wave][xx * 4 + cc],
                            0, 0);
                    }
#else
                    float v = (m >= 0 && m < W_DIM) ? src[m] : 0.0f;
                    sB[bufsel][wave][xx * 4 + cc] = v;
#endif
                }
            }
        }
    };

    v8f acc[6];
#pragma unroll
    for (int nt = 0; nt < 6; ++nt)
        acc[nt] = (v8f){0.f, 0.f, 0.f, 0.f, 0.f, 0.f, 0.f, 0.f};

    stage(0, 0);   // prime buffer 0

    for (int k0 = 0; k0 < C_CH; k0 += KSTEP) {
        const int bufsel = (k0 >> 4) & 1;
#if USE_ASYNC_LDS
        __builtin_amdgcn_s_wait_asynccnt(0);   // my async writes of bufsel done
#endif
        __syncthreads();                        // all waves' staging published

        if (k0 + KSTEP < C_CH)
            stage(bufsel ^ 1, k0 + KSTEP);      // overlap with compute below

        // ---- 4 sub-chunks of K=4, 6 WMMA each ----
#pragma unroll
        for (int sub = 0; sub < 4; ++sub) {
            const int kc = k0 + sub * 4;
            // A fragment (16x4 f32): VGPR0 -> K {0,2}, VGPR1 -> K {1,3}; M = mn
            const float* __restrict__ ap =
                f1 + rowbase + (long)(kc + lhalf * 2) * HW + (w0 + mn);
            v2f A;
            A.x = ap[0];
            A.y = ap[HW];

#pragma unroll
            for (int nt = 0; nt < 6; ++nt) {
                // B fragment (4x16 f32): N = mn, lanes<16 -> K {0,1}, >=16 -> K {2,3}
                const int xl = wave * 16 + nt * 16 + mn;  // block-local f2p col
                const v2f Bf = *(const v2f*)&sB[bufsel][sub][xl * 4 + lhalf * 2];
                acc[nt] = __builtin_amdgcn_wmma_f32_16x16x4_f32(
                    /*neg_a=*/false, A, /*neg_b=*/false, Bf,
                    /*c_mod=*/(short)0, acc[nt],
                    /*reuse_a=*/false, /*reuse_b=*/false);
            }
        }
    }

    // ---- scatter C/D tiles into sOut as [j][w], scaled by 1/C (mean) ----
    __syncthreads();   // compute done everywhere before sOut reuse pattern
    const float scale = 1.0f / (float)C_CH;
#pragma unroll
    for (int nt = 0; nt < 6; ++nt) {
#pragma unroll
        for (int r = 0; r < 8; ++r) {
            const int M = r + lhalf * 8;        // D layout: VGPR r, lane half
            const int j = nt * 16 + mn - M;     // displacement = N_global - M
            if (j >= 0 && j < NJ)
                sOut[j * 65 + wave * 16 + M] = acc[nt][r] * scale;
        }
    }
    __syncthreads();

    // ---- coalesced store: 81 rows x 64 contiguous floats ----
    const long obase = (((long)b * NJ) * H_DIM + h) * W_DIM + wchunk * WCHUNK;
    for (int e = tid; e < NJ * WCHUNK; e += 128) {
        const int j  = e >> 6;
        const int wl = e & 63;
        out[obase + (long)j * (H_DIM * W_DIM) + wl] = sOut[j * 65 + wl];
    }
}

extern "C" void kernel_launch(void* const* d_in, const int* in_sizes, int n_in,
                              void* d_out, int out_size, void* d_ws, size_t ws_size,
                              hipStream_t stream) {
    (void)in_sizes; (void)n_in; (void)d_ws; (void)ws_size; (void)out_size;
    const float* f1 = (const float*)d_in[0];
    const float* f2 = (const float*)d_in[1];
    float* out = (float*)d_out;

    dim3 grid(W_DIM / WCHUNK, H_DIM, 8);   // (5, 96, 8)
    dim3 block(128);
    corr_wmma_kernel<<<grid, block, 0, stream>>>(f1, f2, out);
}